// LSTM_GNN_Feedback_60301340836191
// MI455X (gfx1250) — compile-verified
//
#include <hip/hip_runtime.h>
#include <math.h>

// ---------------------------------------------------------------------------
// LSTM-GNN feedback pipeline for MI455X (gfx1250), fp32 WMMA (16x16x4).
// ---------------------------------------------------------------------------

typedef float v2f __attribute__((ext_vector_type(2)));
typedef float v8f __attribute__((ext_vector_type(8)));

#define NN    100000
#define NE    1600000
#define TT    8
#define DSF   10
#define DTF   20
#define XF    30          // DS + DT
#define GH    64
#define LH    64
#define KIH   84          // DT + GH
#define FUSED 128         // GH + LH
#define ECC   64          // EC
#define EPSV  1e-5f

__device__ __forceinline__ v8f wmma4(v2f a, v2f b, v8f c) {
  // D = A(16x4 f32) * B(4x16 f32) + C(16x16 f32)
  return __builtin_amdgcn_wmma_f32_16x16x4_f32(
      /*neg_a=*/false, a, /*neg_b=*/false, b,
      /*c_mod=*/(short)0, c, /*reuse_a=*/false, /*reuse_b=*/false);
}

__device__ __forceinline__ float sigf(float x) { return 1.0f / (1.0f + __expf(-x)); }

// ---------------------------------------------------------------------------
__global__ void zero_f32(float* __restrict__ p, int n) {
  int i = blockIdx.x * blockDim.x + threadIdx.x;
  if (i < n) p[i] = 0.0f;
}

// ---------------------------------------------------------------------------
// agg[dst] += h[src]  over all edges; one thread per (edge, 4-float chunk)
__global__ void scatter_h(const long long* __restrict__ ei,
                          const float* __restrict__ h,
                          float* __restrict__ agg) {
  long long idx = (long long)blockIdx.x * blockDim.x + threadIdx.x;
  if (idx >= (long long)NE * 16) return;
  int e = (int)(idx >> 4);
  int f = ((int)idx & 15) << 2;
  long long s = ei[e];
  long long d = ei[NE + e];
  const float4 v = *(const float4*)(h + s * GH + f);
  float* a = agg + d * GH + f;
  unsafeAtomicAdd(a + 0, v.x);
  unsafeAtomicAdd(a + 1, v.y);
  unsafeAtomicAdd(a + 2, v.z);
  unsafeAtomicAdd(a + 3, v.w);
}

// ---------------------------------------------------------------------------
// One LSTM/GNN time step. 8 waves per block, 16 nodes per wave.
__global__ __launch_bounds__(256) void lstm_step(
    const float* __restrict__ x, const float* __restrict__ agg,
    float* __restrict__ h, float* __restrict__ c, float* __restrict__ gnn,
    const float* __restrict__ W_rel, const float* __restrict__ b_rel,
    const float* __restrict__ W_root,
    const float* __restrict__ W_ih, const float* __restrict__ W_hh,
    const float* __restrict__ b_ih, const float* __restrict__ b_hh, int t) {
  __shared__ float sg[8 * 16 * 68];  // per-wave 16x64 gnn tile (padded stride 68)

  const int wave = threadIdx.x >> 5;
  const int lane = threadIdx.x & 31;
  const int nm   = lane & 15;          // M for A-frag, N for B/C/D-frag
  const int kh   = (lane >> 4) << 1;   // K half-offset (0 or 2)
  const int tile = blockIdx.x * 8 + wave;
  const int rb   = tile * 16;
  if (rb >= NN) return;

  float* myg = sg + wave * (16 * 68);
  const float* aggR = agg + (long long)rb * GH;
  const float* hR   = h   + (long long)rb * GH;

  // ---- gnn_out = agg @ W_rel^T + b_rel + h @ W_root^T  (16x64)
  for (int nb = 0; nb < GH; nb += 16) {
    v8f acc;
    {
      float bias = b_rel[nb + nm];
      #pragma unroll
      for (int r = 0; r < 8; ++r) acc[r] = bias;
    }
    for (int kk = 0; kk < GH; kk += 4) {
      v2f a, b;
      a.x = aggR[nm * GH + kk + kh];
      a.y = aggR[nm * GH + kk + kh + 1];
      b.x = W_rel[(nb + nm) * GH + kk + kh];
      b.y = W_rel[(nb + nm) * GH + kk + kh + 1];
      acc = wmma4(a, b, acc);
      a.x = hR[nm * GH + kk + kh];
      a.y = hR[nm * GH + kk + kh + 1];
      b.x = W_root[(nb + nm) * GH + kk + kh];
      b.y = W_root[(nb + nm) * GH + kk + kh + 1];
      acc = wmma4(a, b, acc);
    }
    const int ro = (lane >> 4) * 8;
    #pragma unroll
    for (int r = 0; r < 8; ++r) {
      int m = r + ro;
      myg[m * 68 + nb + nm] = acc[r];
      gnn[(long long)(rb + m) * GH + nb + nm] = acc[r];
    }
  }

  // ---- gates = [x_t, gnn] @ W_ih^T + b_ih + h @ W_hh^T + b_hh  (16x256)
  v8f g[16];
  #pragma unroll
  for (int j = 0; j < 16; ++j) {
    float bias = b_ih[j * 16 + nm] + b_hh[j * 16 + nm];
    #pragma unroll
    for (int r = 0; r < 8; ++r) g[j][r] = bias;
  }

  // x_t part (K = 20 -> W_ih cols 0..19); x is (N, 30, T) row-major
  const float* xR = x + (long long)rb * (XF * TT);
  for (int kk = 0; kk < DTF; kk += 4) {
    v2f a;
    a.x = xR[nm * (XF * TT) + (DSF + kk + kh) * TT + t];
    a.y = xR[nm * (XF * TT) + (DSF + kk + kh + 1) * TT + t];
    #pragma unroll
    for (int j = 0; j < 16; ++j) {
      v2f b;
      b.x = W_ih[(j * 16 + nm) * KIH + kk + kh];
      b.y = W_ih[(j * 16 + nm) * KIH + kk + kh + 1];
      g[j] = wmma4(a, b, g[j]);
    }
  }

  // gnn part (K = 64 -> W_ih cols 20..83), A from LDS
  for (int kk = 0; kk < GH; kk += 4) {
    v2f a;
    a.x = myg[nm * 68 + kk + kh];
    a.y = myg[nm * 68 + kk + kh + 1];
    #pragma unroll
    for (int j = 0; j < 16; ++j) {
      v2f b;
      b.x = W_ih[(j * 16 + nm) * KIH + DTF + kk + kh];
      b.y = W_ih[(j * 16 + nm) * KIH + DTF + kk + kh + 1];
      g[j] = wmma4(a, b, g[j]);
    }
  }

  // h part (K = 64, W_hh)
  for (int kk = 0; kk < GH; kk += 4) {
    v2f a;
    a.x = hR[nm * GH + kk + kh];
    a.y = hR[nm * GH + kk + kh + 1];
    #pragma unroll
    for (int j = 0; j < 16; ++j) {
      v2f b;
      b.x = W_hh[(j * 16 + nm) * GH + kk + kh];
      b.y = W_hh[(j * 16 + nm) * GH + kk + kh + 1];
      g[j] = wmma4(a, b, g[j]);
    }
  }

  // ---- nonlinearities + in-place state update
  const int ro = (lane >> 4) * 8;
  #pragma unroll
  for (int jb = 0; jb < 4; ++jb) {
    #pragma unroll
    for (int r = 0; r < 8; ++r) {
      int m = r + ro;
      long long ci = (long long)(rb + m) * LH + jb * 16 + nm;
      float iv = sigf(g[jb][r]);
      float fv = sigf(g[4 + jb][r]);
      float gv = tanhf(g[8 + jb][r]);
      float ov = sigf(g[12 + jb][r]);
      float cn = fv * c[ci] + iv * gv;
      c[ci] = cn;
      h[ci] = ov * tanhf(cn);
    }
  }
}

// ---------------------------------------------------------------------------
// Per-feature mean/var over N for fused = concat([h, gnn]); emits scale/shift.
__global__ void feat_stats(const float* __restrict__ h, const float* __restrict__ gnn,
                           const float* __restrict__ gamma, const float* __restrict__ beta,
                           float* __restrict__ scale, float* __restrict__ shift) {
  __shared__ float ssum[256], ssq[256];
  int j = blockIdx.x;
  const float* p = (j < GH) ? (h + j) : (gnn + (j - GH));
  float s = 0.0f, q = 0.0f;
  for (int n = threadIdx.x; n < NN; n += blockDim.x) {
    float v = p[(long long)n * GH];
    s += v;
    q += v * v;
  }
  ssum[threadIdx.x] = s;
  ssq[threadIdx.x]  = q;
  __syncthreads();
  for (int d = 128; d > 0; d >>= 1) {
    if (threadIdx.x < d) {
      ssum[threadIdx.x] += ssum[threadIdx.x + d];
      ssq[threadIdx.x]  += ssq[threadIdx.x + d];
    }
    __syncthreads();
  }
  if (threadIdx.x == 0) {
    float mean = ssum[0] / (float)NN;
    float var  = ssq[0] / (float)NN - mean * mean;
    float sc   = gamma[j] * rsqrtf(var + EPSV);
    scale[j] = sc;
    shift[j] = beta[j] - mean * sc;
  }
}

// ---------------------------------------------------------------------------
// Batch-norm apply + MLP (128 -> 64 -> 128 -> 1) + sigmoid. 4 waves per block.
__global__ __launch_bounds__(128) void mlp_out(
    const float* __restrict__ h, const float* __restrict__ gnn,
    const float* __restrict__ scale, const float* __restrict__ shift,
    const float* __restrict__ W1, const float* __restrict__ b1,
    const float* __restrict__ W2, const float* __restrict__ b2,
    const float* __restrict__ W_out, const float* __restrict__ b_out,
    float* __restrict__ out) {
  __shared__ float snorm[4 * 16 * 132];
  __shared__ float sh1[4 * 16 * 68];

  const int wave = threadIdx.x >> 5;
  const int lane = threadIdx.x & 31;
  const int nm   = lane & 15;
  const int kh   = (lane >> 4) << 1;
  const int tile = blockIdx.x * 4 + wave;
  const int rb   = tile * 16;
  if (rb >= NN) return;

  float* myn  = snorm + wave * (16 * 132);
  float* myh1 = sh1   + wave * (16 * 68);

  // normalize fused tile into LDS
  for (int idx = lane; idx < 16 * FUSED; idx += 32) {
    int m = idx >> 7, j = idx & 127;
    float v = (j < GH) ? h[(long long)(rb + m) * GH + j]
                       : gnn[(long long)(rb + m) * GH + j - GH];
    myn[m * 132 + j] = v * scale[j] + shift[j];
  }

  // h1 = relu(normed @ W1^T + b1): 16x64, K=128
  for (int nb = 0; nb < ECC; nb += 16) {
    v8f acc;
    {
      float bias = b1[nb + nm];
      #pragma unroll
      for (int r = 0; r < 8; ++r) acc[r] = bias;
    }
    for (int kk = 0; kk < FUSED; kk += 4) {
      v2f a, b;
      a.x = myn[nm * 132 + kk + kh];
      a.y = myn[nm * 132 + kk + kh + 1];
      b.x = W1[(nb + nm) * FUSED + kk + kh];
      b.y = W1[(nb + nm) * FUSED + kk + kh + 1];
      acc = wmma4(a, b, acc);
    }
    const int ro = (lane >> 4) * 8;
    #pragma unroll
    for (int r = 0; r < 8; ++r) {
      int m = r + ro;
      myh1[m * 68 + nb + nm] = fmaxf(acc[r], 0.0f);
    }
  }

  // hidden = relu(h1 @ W2^T + b2): 16x128, K=64; overwrite myn with hidden
  for (int nb = 0; nb < FUSED; nb += 16) {
    v8f acc;
    {
      float bias = b2[nb + nm];
      #pragma unroll
      for (int r = 0; r < 8; ++r) acc[r] = bias;
    }
    for (int kk = 0; kk < ECC; kk += 4) {
      v2f a, b;
      a.x = myh1[nm * 68 + kk + kh];
      a.y = myh1[nm * 68 + kk + kh + 1];
      b.x = W2[(nb + nm) * ECC + kk + kh];
      b.y = W2[(nb + nm) * ECC + kk + kh + 1];
      acc = wmma4(a, b, acc);
    }
    const int ro = (lane >> 4) * 8;
    #pragma unroll
    for (int r = 0; r < 8; ++r) {
      int m = r + ro;
      myn[m * 132 + nb + nm] = fmaxf(acc[r], 0.0f);
    }
  }

  // out = sigmoid(hidden @ W_out^T + b_out): one row per lane 0..15
  if (lane < 16) {
    float acc = b_out[0];
    for (int j = 0; j < FUSED; ++j) acc += myn[lane * 132 + j] * W_out[j];
    out[rb + lane] = sigf(acc);
  }
}

// ---------------------------------------------------------------------------
extern "C" void kernel_launch(void* const* d_in, const int* in_sizes, int n_in,
                              void* d_out, int out_size, void* d_ws, size_t ws_size,
                              hipStream_t stream) {
  const float*     x      = (const float*)d_in[0];
  const long long* ei     = (const long long*)d_in[1];
  // d_in[2], d_in[3] = W_static, b_static: dead code in the reference, unused.
  const float* W_ih  = (const float*)d_in[4];
  const float* W_hh  = (const float*)d_in[5];
  const float* b_ih  = (const float*)d_in[6];
  const float* b_hh  = (const float*)d_in[7];
  const float* W_rel = (const float*)d_in[8];
  const float* b_rel = (const float*)d_in[9];
  const float* W_root = (const float*)d_in[10];
  const float* gamma = (const float*)d_in[11];
  const float* beta  = (const float*)d_in[12];
  const float* W1    = (const float*)d_in[13];
  const float* b1    = (const float*)d_in[14];
  const float* W2    = (const float*)d_in[15];
  const float* b2    = (const float*)d_in[16];
  const float* W_out = (const float*)d_in[17];
  const float* b_out = (const float*)d_in[18];
  float* out = (float*)d_out;

  // Workspace layout (floats): h | c | agg | gnn | scale(128) | shift(128)
  float* h     = (float*)d_ws;
  float* c     = h   + (size_t)NN * GH;
  float* agg   = c   + (size_t)NN * GH;
  float* gnn   = agg + (size_t)NN * GH;
  float* scale = gnn + (size_t)NN * GH;
  float* shift = scale + FUSED;

  {  // h and c are contiguous: zero both in one launch
    int n = NN * GH * 2;
    zero_f32<<<(n + 255) / 256, 256, 0, stream>>>(h, n);
  }

  const int nTiles = (NN + 15) / 16;  // 6250
  for (int t = 0; t < TT; ++t) {
    int n = NN * GH;
    zero_f32<<<(n + 255) / 256, 256, 0, stream>>>(agg, n);
    long long sthreads = (long long)NE * 16;
    scatter_h<<<(unsigned)((sthreads + 255) / 256), 256, 0, stream>>>(ei, h, agg);
    lstm_step<<<(nTiles + 7) / 8, 256, 0, stream>>>(
        x, agg, h, c, gnn, W_rel, b_rel, W_root, W_ih, W_hh, b_ih, b_hh, t);
  }

  feat_stats<<<FUSED, 256, 0, stream>>>(h, gnn, gamma, beta, scale, shift);
  mlp_out<<<(nTiles + 3) / 4, 128, 0, stream>>>(h, gnn, scale, shift,
                                                W1, b1, W2, b2, W_out, b_out, out);
}